// ORB_17059610100466
// MI455X (gfx1250) — compile-verified
//
#include <hip/hip_runtime.h>

typedef float v2f __attribute__((ext_vector_type(2)));
typedef float v8f __attribute__((ext_vector_type(8)));

#define FEAT_D 256

// ---------------------------------------------------------------------------
// Kernel 1: exclusive scan of n_nodes -> row offsets (B <= 4096).
// Single block, 1024 threads, 4 elements/thread, Hillis-Steele in LDS.
// ---------------------------------------------------------------------------
__global__ __launch_bounds__(1024)
void seg_offsets_kernel(const int* __restrict__ n_nodes,
                        int* __restrict__ offsets, int B) {
  __shared__ int sd[1024];
  const int t = threadIdx.x;
  const int b0 = t * 4;
  int v0 = (b0 + 0 < B) ? n_nodes[b0 + 0] : 0;
  int v1 = (b0 + 1 < B) ? n_nodes[b0 + 1] : 0;
  int v2 = (b0 + 2 < B) ? n_nodes[b0 + 2] : 0;
  int v3 = (b0 + 3 < B) ? n_nodes[b0 + 3] : 0;
  const int psum = v0 + v1 + v2 + v3;
  sd[t] = psum;
  __syncthreads();
  for (int o = 1; o < 1024; o <<= 1) {
    int x = (t >= o) ? sd[t - o] : 0;
    __syncthreads();
    sd[t] += x;
    __syncthreads();
  }
  const int excl = sd[t] - psum;                 // exclusive prefix for this group
  if (b0 + 0 < B) offsets[b0 + 0] = excl;
  if (b0 + 1 < B) offsets[b0 + 1] = excl + v0;
  if (b0 + 2 < B) offsets[b0 + 2] = excl + v0 + v1;
  if (b0 + 3 < B) offsets[b0 + 3] = excl + v0 + v1 + v2;
}

// ---------------------------------------------------------------------------
// Kernel 2: segment mean via V_WMMA_F32_16X16X4_F32 ones-vector reduction.
// Grid: one block per graph (4096). Block: 512 threads = 16 waves.
// Wave w owns columns [16w, 16w+16). Per WMMA: 4 rows x 16 cols reduced.
//
// A (16x4 f32, 2 VGPRs): lanes 0-15 hold M=lane (K0 in v0, K1 in v1);
// lanes 16-31 hold M=lane-16 (K2 in v0, K3 in v1). Row M=0 = ones -> D row 0
// accumulates the 4-row column sum; rows 1..15 of C are passed through.
//
// B (4x16 f32, 2 VGPRs): v0 = row k (lanes 0-15) / row k+2 (lanes 16-31);
// v1 = row k+1 / row k+3; N = lane & 15.
// ---------------------------------------------------------------------------
__global__ __launch_bounds__(512)
void seg_mean_wmma_kernel(const float* __restrict__ feats,
                          const int* __restrict__ n_nodes,
                          const int* __restrict__ offsets,
                          float* __restrict__ out) {
  const int g      = blockIdx.x;
  const int lane   = threadIdx.x & 31;
  const int wave   = threadIdx.x >> 5;
  const int nIdx   = lane & 15;
  const int col    = (wave << 4) + nIdx;       // column this lane produces
  const int rowOff = (lane >> 4) << 1;         // 0 for lanes 0-15, 2 for 16-31

  const int s   = offsets[g];
  const int n   = n_nodes[g];
  const int end = s + n;

  v2f a;
  a.x = (nIdx == 0) ? 1.0f : 0.0f;             // A row M=0 = ones, rest zero
  a.y = a.x;

  v8f c0 = {};
  v8f c1 = {};

  int r = s;
  // Steady state: 8 rows / iteration, two independent accumulator chains.
  for (; r + 8 <= end; r += 8) {
    const float* p0 = feats + (size_t)(r + rowOff) * FEAT_D + col;
    v2f b0; b0.x = p0[0];            b0.y = p0[FEAT_D];
    const float* p1 = p0 + 4 * FEAT_D;
    v2f b1; b1.x = p1[0];            b1.y = p1[FEAT_D];
    __builtin_prefetch(p0 + 16 * FEAT_D, 0, 0);   // ~4 rows x 4 ahead
    c0 = __builtin_amdgcn_wmma_f32_16x16x4_f32(false, a, false, b0,
                                               (short)0, c0, false, false);
    c1 = __builtin_amdgcn_wmma_f32_16x16x4_f32(false, a, false, b1,
                                               (short)0, c1, false, false);
  }
  // Tail: up to 7 rows, zero-masked lanes keep the reduction exact.
  for (; r < end; r += 4) {
    const int rx = r + rowOff;
    v2f b0;
    b0.x = (rx     < end) ? feats[(size_t)rx       * FEAT_D + col] : 0.0f;
    b0.y = (rx + 1 < end) ? feats[(size_t)(rx + 1) * FEAT_D + col] : 0.0f;
    c0 = __builtin_amdgcn_wmma_f32_16x16x4_f32(false, a, false, b0,
                                               (short)0, c0, false, false);
  }

  if (lane < 16) {                              // D row 0 lives in c[0], lanes 0-15
    const float sum = c0[0] + c1[0];
    const float inv = (n > 0) ? (1.0f / (float)n) : 0.0f;
    out[(size_t)g * FEAT_D + col] = sum * inv;
  }
}

// ---------------------------------------------------------------------------
// Launch
// ---------------------------------------------------------------------------
extern "C" void kernel_launch(void* const* d_in, const int* in_sizes, int n_in,
                              void* d_out, int out_size, void* d_ws, size_t ws_size,
                              hipStream_t stream) {
  const float* feats   = (const float*)d_in[0];
  const int*   n_nodes = (const int*)d_in[1];
  const int    B       = in_sizes[1];

  int* offsets = (int*)d_ws;                    // B ints of scratch

  seg_offsets_kernel<<<1, 1024, 0, stream>>>(n_nodes, offsets, B);
  seg_mean_wmma_kernel<<<B, 512, 0, stream>>>(feats, n_nodes, offsets,
                                              (float*)d_out);
}